// LSS_15367392985751
// MI455X (gfx1250) — compile-verified
//
#include <hip/hip_runtime.h>
#include <math.h>

typedef float v2f __attribute__((ext_vector_type(2)));
typedef float v8f __attribute__((ext_vector_type(8)));

#define B_   2
#define N_   6
#define C_   64
#define D_   41
#define H_   32
#define W_   88
#define HW_  (H_ * W_)          // 2816
#define NX0  200
#define NX1  200
#define NPIX (B_ * N_ * H_ * W_) // 33792
#define WAVES_PER_BLOCK 8

// voxel flat offset (ix*200+iy) within one batch slice, or -1 if invalid.
// Mirrors: world = geom*EXTENT + LO ; vox = floor((world - LO)/DX)
__device__ __forceinline__ int vox_offset(float gx, float gy, float gz) {
    float wx = fmaf(gx, 100.0f, -50.0f);
    float wy = fmaf(gy, 100.0f, -50.0f);
    float wz = fmaf(gz,  20.0f, -10.0f);
    int ix = (int)floorf((wx + 50.0f) / 0.5f);
    int iy = (int)floorf((wy + 50.0f) / 0.5f);
    int iz = (int)floorf((wz + 10.0f) / 20.0f);
    bool valid = (ix >= 0) && (ix < NX0) && (iy >= 0) && (iy < NX1) && (iz == 0);
    return valid ? (ix * NX1 + iy) : -1;
}

// One wave (32 lanes) handles one pixel (b,n,h,w):
//   softmax over D=41, then the 41x64 outer product via 12x
//   v_wmma_f32_16x16x4_f32 (K=1 payload in k=0, rows k=1..3 zero),
//   then scatter-add into the BEV grid with global f32 atomics (hit L2).
__global__ __launch_bounds__(32 * WAVES_PER_BLOCK)
void lss_bevpool_wmma(const float* __restrict__ feat,
                      const float* __restrict__ dlog,
                      const float* __restrict__ geom,
                      float* __restrict__ out) {
    __shared__ float s_dep[WAVES_PER_BLOCK][48];
    __shared__ float s_ft [WAVES_PER_BLOCK][64];
    __shared__ int   s_off[WAVES_PER_BLOCK][48];

    const int lane = threadIdx.x & 31;
    const int wid  = threadIdx.x >> 5;
    const int pix  = blockIdx.x * WAVES_PER_BLOCK + wid;   // exact cover: no tail

    const int hw = pix % HW_;
    const int bn = pix / HW_;
    const int b  = bn / N_;

    const float* fptr = feat + (size_t)bn * C_ * HW_ + hw;
    const float* dptr = dlog + (size_t)bn * D_ * HW_ + hw;
    const float* gptr = geom + ((size_t)bn * D_ * HW_ + hw) * 3;

    // ---------------- softmax over D = 41 ----------------
    float x0 = dptr[lane * HW_];
    float x1 = (lane < (D_ - 32)) ? dptr[(lane + 32) * HW_] : -3.402823466e38f;
    float m = fmaxf(x0, x1);
    #pragma unroll
    for (int o = 16; o > 0; o >>= 1) m = fmaxf(m, __shfl_xor(m, o, 32));
    float e0 = __expf(x0 - m);
    float e1 = (lane < (D_ - 32)) ? __expf(x1 - m) : 0.0f;
    float s = e0 + e1;
    #pragma unroll
    for (int o = 16; o > 0; o >>= 1) s += __shfl_xor(s, o, 32);
    float inv = 1.0f / s;

    s_dep[wid][lane] = e0 * inv;
    if (lane < 16) s_dep[wid][lane + 32] = e1 * inv;  // lanes 9..15 write 0 pad

    // ---------------- feat (64 channels, stride HW) ----------------
    s_ft[wid][lane]      = fptr[lane * HW_];
    s_ft[wid][lane + 32] = fptr[(lane + 32) * HW_];

    // ---------------- per-d voxel offsets ----------------
    {
        const float* g = gptr + (size_t)lane * 3 * HW_;
        s_off[wid][lane] = vox_offset(g[0], g[1], g[2]);
    }
    if (lane < 16) {
        int o2 = -1;
        if (lane < (D_ - 32)) {
            const float* g = gptr + (size_t)(lane + 32) * 3 * HW_;
            o2 = vox_offset(g[0], g[1], g[2]);
        }
        s_off[wid][lane + 32] = o2;
    }
    __syncthreads();

    const float* dep  = s_dep[wid];
    const float* ft   = s_ft[wid];
    const int*   voff = s_off[wid];
    float* obase = out + (size_t)b * (C_ * NX0 * NX1);

    const int  nsel = lane & 15;
    const bool lo16 = (lane < 16);
    const int  mrow = (lane >= 16) ? 8 : 0;

    #pragma unroll
    for (int mt = 0; mt < 3; ++mt) {
        // A fragment 16x4: k=0 row carries depth tile, k=1..3 zero
        v2f a;
        a.x = lo16 ? dep[mt * 16 + lane] : 0.0f;
        a.y = 0.0f;

        // voxel offsets for the 8 C/D rows this lane owns in this M tile
        int off[8];
        #pragma unroll
        for (int v = 0; v < 8; ++v) off[v] = voff[mt * 16 + v + mrow];

        #pragma unroll
        for (int nt = 0; nt < 4; ++nt) {
            // B fragment 4x16: k=0 row carries feat tile, k=1..3 zero
            v2f bb;
            bb.x = lo16 ? ft[nt * 16 + lane] : 0.0f;
            bb.y = 0.0f;

            v8f acc = {};
            acc = __builtin_amdgcn_wmma_f32_16x16x4_f32(
                false, a, false, bb, (short)0, acc, false, false);

            float* ochan = obase + (size_t)(nt * 16 + nsel) * (NX0 * NX1);
            #pragma unroll
            for (int v = 0; v < 8; ++v) {
                int o = off[v];
                if (o >= 0) atomicAdd(ochan + o, acc[v]);
            }
        }
    }
}

extern "C" void kernel_launch(void* const* d_in, const int* in_sizes, int n_in,
                              void* d_out, int out_size, void* d_ws, size_t ws_size,
                              hipStream_t stream) {
    const float* feat = (const float*)d_in[0];
    const float* dlog = (const float*)d_in[1];
    const float* geom = (const float*)d_in[2];
    float* out = (float*)d_out;

    // Zero the BEV accumulator (graph-capturable memset node)
    hipMemsetAsync(out, 0, (size_t)out_size * sizeof(float), stream);

    const int blocks = NPIX / WAVES_PER_BLOCK;   // 33792 / 8 = 4224, exact
    lss_bevpool_wmma<<<blocks, 32 * WAVES_PER_BLOCK, 0, stream>>>(feat, dlog, geom, out);
}